// MeanShiftLayer_15341623181313
// MI455X (gfx1250) — compile-verified
//
#include <hip/hip_runtime.h>
#include <stdint.h>

// Problem constants (from reference)
#define N_PTS   200000
#define DIM     64
#define K_C     256
#define BW2     144.0f        // bandwidth^2
#define ITERS   10

#define TILE_N  512                        // points per block
#define NGRP    (TILE_N / 16)              // 32 point-groups per block
#define NBLK    ((N_PTS + TILE_N - 1) / TILE_N)   // 391
#define XSTRIDE (DIM + 4)                  // LDS row stride (bank-conflict padding, keeps 16B align)

typedef float v2f __attribute__((ext_vector_type(2)));
typedef float v8f __attribute__((ext_vector_type(8)));

__device__ __forceinline__ v8f wmma_f32(v2f a, v2f b, v8f c) {
  // D(16x16,f32) = A(16x4,f32) * B(4x16,f32) + C
  return __builtin_amdgcn_wmma_f32_16x16x4_f32(
      /*neg_a=*/false, a, /*neg_b=*/false, b,
      /*c_mod=*/(short)0, c, /*reuse_a=*/false, /*reuse_b=*/false);
}

// Async global->LDS staging of one TILE_N x DIM chunk of x (128-bit chunks),
// then compute x^2 per row from LDS.  Runs with all 256 threads.
template <typename XS>
__device__ __forceinline__ void stage_x_async(const float* __restrict__ x,
                                              XS& xs, float* x2s,
                                              int pbase, int tid) {
  // 512 rows * 16 float4-chunks = 8192 chunks; 32 per thread
  for (int i = 0; i < (TILE_N * 16) / 256; ++i) {
    const int cid = tid + i * 256;
    const int p   = cid >> 4;       // row in tile
    const int j   = cid & 15;       // float4 index within row
    const int gp  = pbase + p;
    if (gp < N_PTS) {
      const float*  gptr   = x + (size_t)gp * DIM + j * 4;
      const unsigned ldsoff =
          (unsigned)(uintptr_t)(&xs[p][j * 4]);   // low 32 bits = LDS offset
      asm volatile("global_load_async_to_lds_b128 %0, %1, off"
                   :: "v"(ldsoff), "v"(gptr) : "memory");
    } else {
      ((float4*)&xs[p][j * 4])[0] = make_float4(0.f, 0.f, 0.f, 0.f);
    }
  }
  asm volatile("s_wait_asynccnt 0" ::: "memory");
  __syncthreads();
  // x^2 per row (2 rows per thread), read back from LDS as float4
  for (int rr = 0; rr < 2; ++rr) {
    const int p = tid * 2 + rr;
    const float4* rp = (const float4*)&xs[p][0];
    float s = 0.0f;
    for (int j = 0; j < DIM / 4; ++j) {
      const float4 v = rp[j];
      s += v.x * v.x + v.y * v.y + v.z * v.z + v.w * v.w;
    }
    x2s[p] = (pbase + p < N_PTS) ? s : 3.0e38f;  // invalid rows never pass mask
  }
  __syncthreads();
}

// ---------------------------------------------------------------------------
// init: gather centers = x[seed], compute c2.  grid = K_C blocks of 64
// ---------------------------------------------------------------------------
__global__ __launch_bounds__(64)
void ms_init_kernel(const float* __restrict__ x, const int* __restrict__ seed,
                    float* __restrict__ centers, float* __restrict__ c2) {
  const int k = blockIdx.x;
  const int d = threadIdx.x;
  __shared__ float red[64];
  const int s = seed[k];
  const float v = x[(size_t)s * DIM + d];
  centers[k * DIM + d] = v;
  red[d] = v * v;
  __syncthreads();
  for (int off = 32; off >= 1; off >>= 1) {
    if (d < off) red[d] += red[d + off];
    __syncthreads();
  }
  if (d == 0) c2[k] = red[0];
}

// ---------------------------------------------------------------------------
// zero accumulators + argmin keys.  grid = 64 blocks of 256
// ---------------------------------------------------------------------------
__global__ __launch_bounds__(256)
void ms_zero_kernel(float* __restrict__ sums, float* __restrict__ counts,
                    unsigned long long* __restrict__ keys) {
  const int i = blockIdx.x * 256 + threadIdx.x;
  if (i < K_C * DIM) sums[i] = 0.0f;
  if (i < K_C) {
    counts[i] = 0.0f;
    // key = (bits(+inf) << 32) | 0  => empty mask yields argmin index 0
    keys[i] = 0x7F80000000000000ull;
  }
}

// ---------------------------------------------------------------------------
// main accumulation: distances via WMMA, mask, masked-sum via WMMA, atomics
// grid = NBLK blocks of 256 (8 waves); wave w owns center tiles {2w, 2w+1}
// ---------------------------------------------------------------------------
__global__ __launch_bounds__(256)
void ms_accum_kernel(const float* __restrict__ x,
                     const float* __restrict__ centers,
                     const float* __restrict__ c2,
                     float* __restrict__ sums,
                     float* __restrict__ counts) {
  __shared__ __align__(16) float xs[TILE_N][XSTRIDE];  // staged points
  __shared__ float x2s[TILE_N];
  __shared__ float maskbuf[8][16][17];                 // per-wave transpose buffer

  const int tid   = threadIdx.x;
  const int lane  = tid & 31;
  const int wave  = tid >> 5;
  const int lhalf = lane >> 4;   // 0 | 1
  const int l16   = lane & 15;
  const int pbase = blockIdx.x * TILE_N;

  stage_x_async(x, xs, x2s, pbase, tid);

  for (int t = wave * 2; t < wave * 2 + 2; ++t) {
    const int cbase = t * 16;

    // A fragments (centers, 16x4 tiles): lane holds M=l16, K = 4k + j + 2*lhalf
    v2f afr[16];
    {
      const float* crow = centers + (size_t)(cbase + l16) * DIM + 2 * lhalf;
      for (int k = 0; k < 16; ++k) {
        afr[k].x = crow[4 * k + 0];
        afr[k].y = crow[4 * k + 1];
      }
    }
    float c2f_[8];
    for (int r = 0; r < 8; ++r) c2f_[r] = c2[cbase + r + 8 * lhalf];

    v8f acc0 = {}, acc1 = {}, acc2 = {}, acc3 = {};
    float cnt[8];
    for (int r = 0; r < 8; ++r) cnt[r] = 0.0f;

    for (int g = 0; g < NGRP; ++g) {
      const int prow = g * 16;

      // ---- GEMM 1: dot[center][point]; two independent 8-deep chains ----
      v8f dotA = {}, dotB = {};
      for (int k = 0; k < 8; ++k) {
        v2f bA, bB;   // B(4x16): lane holds N=l16(point), K = j + 2*lhalf (dim)
        bA.x = xs[prow + l16][4 * k + 0 + 2 * lhalf];
        bA.y = xs[prow + l16][4 * k + 1 + 2 * lhalf];
        bB.x = xs[prow + l16][4 * (k + 8) + 0 + 2 * lhalf];
        bB.y = xs[prow + l16][4 * (k + 8) + 1 + 2 * lhalf];
        dotA = wmma_f32(afr[k],     bA, dotA);
        dotB = wmma_f32(afr[k + 8], bB, dotB);
      }

      // ---- flat-kernel mask (D layout: vgpr r -> center r+8*lhalf, lane -> point l16)
      const float px2 = x2s[prow + l16];
      float m[8];
      for (int r = 0; r < 8; ++r) {
        const float d2 = fmaf(-2.0f, dotA[r] + dotB[r], c2f_[r] + px2);
        m[r] = (d2 < BW2) ? 1.0f : 0.0f;
        cnt[r] += m[r];
      }

      // ---- transpose mask tile through wave-private LDS (D layout -> A layout)
      for (int r = 0; r < 8; ++r) maskbuf[wave][r + 8 * lhalf][l16] = m[r];
      asm volatile("s_wait_dscnt 0" ::: "memory");

      // ---- GEMM 2: acc[center][dim] += mask[16x16] @ x[16x64], K-loop over points
      for (int kk = 0; kk < 4; ++kk) {
        v2f am;  // A(16x4): lane holds M=l16(center), K = 4kk + j + 2*lhalf (point)
        am.x = maskbuf[wave][l16][4 * kk + 0 + 2 * lhalf];
        am.y = maskbuf[wave][l16][4 * kk + 1 + 2 * lhalf];
        const int p0 = prow + 4 * kk + 2 * lhalf;
        v2f b0, b1, b2, b3;  // B(4x16): lane holds N=l16(dim), K = j+2*lhalf (point)
        b0.x = xs[p0 + 0][ 0 + l16]; b0.y = xs[p0 + 1][ 0 + l16];
        b1.x = xs[p0 + 0][16 + l16]; b1.y = xs[p0 + 1][16 + l16];
        b2.x = xs[p0 + 0][32 + l16]; b2.y = xs[p0 + 1][32 + l16];
        b3.x = xs[p0 + 0][48 + l16]; b3.y = xs[p0 + 1][48 + l16];
        acc0 = wmma_f32(am, b0, acc0);
        acc1 = wmma_f32(am, b1, acc1);
        acc2 = wmma_f32(am, b2, acc2);
        acc3 = wmma_f32(am, b3, acc3);
      }
    }

    // ---- fold block-local partials into global accumulators ----
    for (int r = 0; r < 8; ++r) {
      const int c = cbase + r + 8 * lhalf;
      atomicAdd(&sums[(size_t)c * DIM +  0 + l16], acc0[r]);
      atomicAdd(&sums[(size_t)c * DIM + 16 + l16], acc1[r]);
      atomicAdd(&sums[(size_t)c * DIM + 32 + l16], acc2[r]);
      atomicAdd(&sums[(size_t)c * DIM + 48 + l16], acc3[r]);
      atomicAdd(&counts[c], cnt[r]);
    }
  }
}

// ---------------------------------------------------------------------------
// new centers = sums / counts; recompute c2.  grid = K_C blocks of 64
// ---------------------------------------------------------------------------
__global__ __launch_bounds__(64)
void ms_divide_kernel(const float* __restrict__ sums,
                      const float* __restrict__ counts,
                      float* __restrict__ centers, float* __restrict__ c2) {
  const int k = blockIdx.x;
  const int d = threadIdx.x;
  __shared__ float red[64];
  const float v = sums[k * DIM + d] / counts[k];
  centers[k * DIM + d] = v;
  red[d] = v * v;
  __syncthreads();
  for (int off = 32; off >= 1; off >>= 1) {
    if (d < off) red[d] += red[d + off];
    __syncthreads();
  }
  if (d == 0) c2[k] = red[0];
}

// ---------------------------------------------------------------------------
// final: within last mask (vs c9), argmin distance to c10 via u64 atomicMin
// ---------------------------------------------------------------------------
__global__ __launch_bounds__(256)
void ms_argmin_kernel(const float* __restrict__ x,
                      const float* __restrict__ c9,  const float* __restrict__ c2_9,
                      const float* __restrict__ c10, const float* __restrict__ c2_10,
                      unsigned long long* __restrict__ keys) {
  __shared__ __align__(16) float xs[TILE_N][XSTRIDE];
  __shared__ float x2s[TILE_N];

  const int tid   = threadIdx.x;
  const int lane  = tid & 31;
  const int wave  = tid >> 5;
  const int lhalf = lane >> 4;
  const int l16   = lane & 15;
  const int pbase = blockIdx.x * TILE_N;

  stage_x_async(x, xs, x2s, pbase, tid);

  for (int t = wave * 2; t < wave * 2 + 2; ++t) {
    const int cbase = t * 16;
    v2f a9[16], a10[16];
    {
      const float* r9  = c9  + (size_t)(cbase + l16) * DIM + 2 * lhalf;
      const float* r10 = c10 + (size_t)(cbase + l16) * DIM + 2 * lhalf;
      for (int k = 0; k < 16; ++k) {
        a9[k].x  = r9[4 * k + 0];  a9[k].y  = r9[4 * k + 1];
        a10[k].x = r10[4 * k + 0]; a10[k].y = r10[4 * k + 1];
      }
    }
    float c2f9[8], c2f10[8];
    for (int r = 0; r < 8; ++r) {
      c2f9[r]  = c2_9[cbase + r + 8 * lhalf];
      c2f10[r] = c2_10[cbase + r + 8 * lhalf];
    }

    unsigned long long best[8];
    for (int r = 0; r < 8; ++r) best[r] = 0x7F80000000000000ull;

    for (int g = 0; g < NGRP; ++g) {
      const int prow = g * 16;
      v8f dot9 = {}, dot10 = {};   // two independent chains interleave in the pipe
      for (int k = 0; k < 16; ++k) {
        v2f b;
        b.x = xs[prow + l16][4 * k + 0 + 2 * lhalf];
        b.y = xs[prow + l16][4 * k + 1 + 2 * lhalf];
        dot9  = wmma_f32(a9[k],  b, dot9);
        dot10 = wmma_f32(a10[k], b, dot10);
      }
      const float px2 = x2s[prow + l16];
      const unsigned pidx = (unsigned)(pbase + prow + l16);
      for (int r = 0; r < 8; ++r) {
        const float d2_9  = fmaf(-2.0f, dot9[r],  c2f9[r]  + px2);
        const float d2_10 = fmaxf(fmaf(-2.0f, dot10[r], c2f10[r] + px2), 0.0f);
        const unsigned long long key =
            ((unsigned long long)__float_as_uint(d2_10) << 32) | pidx;
        const unsigned long long cand =
            (d2_9 < BW2) ? key : 0xFFFFFFFFFFFFFFFFull;
        best[r] = (cand < best[r]) ? cand : best[r];
      }
    }
    for (int r = 0; r < 8; ++r)
      atomicMin(&keys[cbase + r + 8 * lhalf], best[r]);
  }
}

// ---------------------------------------------------------------------------
// pack outputs: [centers | x[argmin] | argmin-as-float]
// ---------------------------------------------------------------------------
__global__ __launch_bounds__(256)
void ms_output_kernel(const float* __restrict__ c10,
                      const unsigned long long* __restrict__ keys,
                      const float* __restrict__ x, float* __restrict__ out) {
  const int i = blockIdx.x * 256 + threadIdx.x;
  if (i < K_C * DIM) {
    out[i] = c10[i];
    const int k = i / DIM, d = i % DIM;
    const unsigned idx = (unsigned)(keys[k] & 0xFFFFFFFFull);
    out[K_C * DIM + i] = x[(size_t)idx * DIM + d];
    if (d == 0) out[2 * K_C * DIM + k] = (float)idx;
  }
}

// ---------------------------------------------------------------------------
extern "C" void kernel_launch(void* const* d_in, const int* in_sizes, int n_in,
                              void* d_out, int out_size, void* d_ws, size_t ws_size,
                              hipStream_t stream) {
  const float* x    = (const float*)d_in[0];
  const int*   seed = (const int*)d_in[1];
  float*       out  = (float*)d_out;

  float* ws = (float*)d_ws;
  float* cA     = ws;                 // K*D
  float* cB     = cA + K_C * DIM;     // K*D
  float* c2A    = cB + K_C * DIM;     // K
  float* c2B    = c2A + K_C;          // K
  float* sums   = c2B + K_C;          // K*D
  float* counts = sums + K_C * DIM;   // K
  unsigned long long* keys = (unsigned long long*)(counts + K_C);  // K (8B aligned)

  ms_init_kernel<<<K_C, 64, 0, stream>>>(x, seed, cA, c2A);

  float *cur = cA, *c2cur = c2A, *nxt = cB, *c2nxt = c2B;
  for (int it = 0; it < ITERS; ++it) {
    ms_zero_kernel<<<(K_C * DIM + 255) / 256, 256, 0, stream>>>(sums, counts, keys);
    ms_accum_kernel<<<NBLK, 256, 0, stream>>>(x, cur, c2cur, sums, counts);
    ms_divide_kernel<<<K_C, 64, 0, stream>>>(sums, counts, nxt, c2nxt);
    float* t0 = cur; cur = nxt; nxt = t0;
    float* t1 = c2cur; c2cur = c2nxt; c2nxt = t1;
  }
  // cur = centers_10, nxt = centers_9 (mask source)
  ms_argmin_kernel<<<NBLK, 256, 0, stream>>>(x, nxt, c2nxt, cur, c2cur, keys);
  ms_output_kernel<<<(K_C * DIM + 255) / 256, 256, 0, stream>>>(cur, keys, x, out);
}